// CLOSEgaps_20950850469932
// MI455X (gfx1250) — compile-verified
//
#include <hip/hip_runtime.h>
#include <math.h>

typedef __attribute__((ext_vector_type(2))) float v2f;
typedef __attribute__((ext_vector_type(8))) float v8f;

#define NN   8192
#define MM   8192
#define FIN  512
#define EMB  256
#define CD   128
#define NH   3
#define DEG  16
#define HFD  (NH * CD)   // 384
#define EE   (MM * DEG)  // 131072

// ---------------------------------------------------------------------------
// f32 WMMA GEMM: C[M,Ncols] = act(A[M,K] @ B[K,Ncols] + bias)
// Register-blocked: each wave computes a 64x32 macro-tile of C as 4x2 WMMA
// 16x16 tiles (V_WMMA_F32_16X16X4_F32, full f32 precision). Per K-step of 4:
// 4 A-fragments + 2 B-fragments feed 8 wmmas -> ~10.7 FLOP/byte of loads.
// Launch: total waves = (M/64)*(Ncols/32), 8 waves (256 thr) per block.
// ---------------------------------------------------------------------------
__global__ __launch_bounds__(256)
void gemm_wmma_f32(const float* __restrict__ A, const float* __restrict__ B,
                   const float* __restrict__ bias, float* __restrict__ C,
                   int K, int Ncols, int relu)
{
    const int wave = threadIdx.x >> 5;
    const int lane = threadIdx.x & 31;
    const int r    = lane & 15;   // A-row within tile / B,C col within tile
    const int kh   = lane >> 4;   // K-half select
    const int NB   = Ncols >> 5;  // number of 32-wide N blocks
    const int wid  = blockIdx.x * 8 + wave;
    const int baseM = (wid / NB) * 64;
    const int baseN = (wid % NB) * 32;

    const float* Ar0 = A + (size_t)(baseM + r) * K;
    const float* Ar1 = Ar0 + (size_t)16 * K;
    const float* Ar2 = Ar1 + (size_t)16 * K;
    const float* Ar3 = Ar2 + (size_t)16 * K;
    const float* Bb  = B + baseN + r;

    v8f acc[4][2];
#pragma unroll
    for (int i = 0; i < 4; i++)
#pragma unroll
        for (int j = 0; j < 2; j++) acc[i][j] = (v8f){};

    for (int k = 0; k < K; k += 4) {
        const int kk = k + 2 * kh;
        v2f av[4], bv[2];
        av[0].x = Ar0[kk]; av[0].y = Ar0[kk + 1];
        av[1].x = Ar1[kk]; av[1].y = Ar1[kk + 1];
        av[2].x = Ar2[kk]; av[2].y = Ar2[kk + 1];
        av[3].x = Ar3[kk]; av[3].y = Ar3[kk + 1];
        const float* Bp = Bb + (size_t)kk * Ncols;
        bv[0].x = Bp[0];  bv[0].y = Bp[Ncols];
        bv[1].x = Bp[16]; bv[1].y = Bp[16 + Ncols];
        // speculative prefetch of the B panel 16 K-rows ahead (OOB is dropped)
        __builtin_prefetch(Bp + (size_t)16 * Ncols, 0, 1);
#pragma unroll
        for (int i = 0; i < 4; i++)
#pragma unroll
            for (int j = 0; j < 2; j++)
                acc[i][j] = __builtin_amdgcn_wmma_f32_16x16x4_f32(
                    false, av[i], false, bv[j], (short)0, acc[i][j], false, false);
    }

#pragma unroll
    for (int i = 0; i < 4; i++)
#pragma unroll
        for (int j = 0; j < 2; j++)
#pragma unroll
            for (int v = 0; v < 8; v++) {
                const int row = baseM + 16 * i + v + 8 * kh;
                const int col = baseN + 16 * j + r;
                float val = acc[i][j][v];
                if (bias) val += bias[col];
                if (relu) val = fmaxf(val, 0.0f);
                C[(size_t)row * Ncols + col] = val;
            }
}

// hyperedge_attr[m,:] = b_attr + sum over 16 incident nodes of W_attr[node,:]
__global__ __launch_bounds__(EMB)
void hedge_attr_kernel(const int* __restrict__ node_idx,
                       const float* __restrict__ W_attr,
                       const float* __restrict__ b_attr,
                       float* __restrict__ hattr)
{
    const int m = blockIdx.x, t = threadIdx.x;
    __shared__ int nd[DEG];
    if (t < DEG) nd[t] = node_idx[m * DEG + t];
    __syncthreads();
    float s = b_attr[t];
#pragma unroll
    for (int j = 0; j < DEG; j++) s += W_attr[(size_t)nd[j] * EMB + t];
    hattr[(size_t)m * EMB + t] = s;
}

// xdot[n,h] = <xl[n,h,:], att[h,:CD]> ; edot[m,h] = <ea[m,h,:], att[h,CD:]>
__global__ void att_dots_kernel(const float* __restrict__ xl,
                                const float* __restrict__ ea,
                                const float* __restrict__ att,
                                float* __restrict__ xdot,
                                float* __restrict__ edot)
{
    const int idx = blockIdx.x * blockDim.x + threadIdx.x;
    if (idx >= NN * NH) return;
    const int n = idx / NH, h = idx % NH;
    const float* a1 = att + h * (2 * CD);
    const float* a2 = a1 + CD;
    const float* xr = xl + (size_t)n * HFD + h * CD;
    const float* er = ea + (size_t)n * HFD + h * CD;
    float s1 = 0.f, s2 = 0.f;
#pragma unroll 4
    for (int f = 0; f < CD; f++) { s1 += xr[f] * a1[f]; s2 += er[f] * a2[f]; }
    xdot[idx] = s1;
    edot[idx] = s2;
}

// node degrees (integer atomics)
__global__ void node_deg_kernel(const int* __restrict__ node_idx,
                                int* __restrict__ degn)
{
    const int e = blockIdx.x * blockDim.x + threadIdx.x;
    if (e < EE) atomicAdd(&degn[node_idx[e]], 1);
}

// per-edge softmax over its 16 entries + node->edge aggregation (Bv = 1/16)
__global__ __launch_bounds__(HFD)
void edge_softmax_eout_kernel(const int* __restrict__ node_idx,
                              const float* __restrict__ xdot,
                              const float* __restrict__ edot,
                              const float* __restrict__ xl,
                              float* __restrict__ alpha_g,
                              float* __restrict__ eout)
{
    const int m = blockIdx.x, t = threadIdx.x;
    __shared__ int nd[DEG];
    __shared__ float al[DEG][NH];
    if (t < DEG) nd[t] = node_idx[m * DEG + t];
    __syncthreads();
    if (t < NH) {
        const int h = t;
        const float ed = edot[m * NH + h];
        float a[DEG];
        float mx = -1e30f;
#pragma unroll
        for (int j = 0; j < DEG; j++) {
            float v = xdot[nd[j] * NH + h] + ed;
            v = (v > 0.f) ? v : 0.2f * v;      // leaky_relu(0.2)
            a[j] = v;
            mx = fmaxf(mx, v);
        }
        float s = 0.f;
#pragma unroll
        for (int j = 0; j < DEG; j++) { a[j] = expf(a[j] - mx); s += a[j]; }
        const float inv = 1.f / (s + 1e-16f);
#pragma unroll
        for (int j = 0; j < DEG; j++) al[j][h] = a[j] * inv;
    }
    __syncthreads();
    if (t < DEG * NH) {
        const int j = t % DEG, h = t / DEG;
        alpha_g[(m * DEG + j) * NH + h] = al[j][h];
    }
    const int h = t / CD;
    float s = 0.f;
#pragma unroll
    for (int j = 0; j < DEG; j++) s += al[j][h] * xl[(size_t)nd[j] * HFD + t];
    eout[(size_t)m * HFD + t] = s * (1.0f / DEG);
}

// edge->node scatter with native f32 atomics; eout row staged once per block
__global__ __launch_bounds__(HFD)
void scatter_nodes_kernel(const int* __restrict__ node_idx,
                          const float* __restrict__ alpha_g,
                          const float* __restrict__ eout,
                          float* __restrict__ nout)
{
    const int m = blockIdx.x, t = threadIdx.x;
    __shared__ int nd[DEG];
    __shared__ float al[DEG][NH];
    if (t < DEG) nd[t] = node_idx[m * DEG + t];
    if (t < DEG * NH) {
        const int j = t % DEG, h = t / DEG;
        al[j][h] = alpha_g[(m * DEG + j) * NH + h];
    }
    __syncthreads();
    const int h = t / CD;
    const float ev = eout[(size_t)m * HFD + t];
#pragma unroll
    for (int j = 0; j < DEG; j++)
        unsafeAtomicAdd(&nout[(size_t)nd[j] * HFD + t], al[j][h] * ev);
}

// conv_out = nout * Dv(n) + b_conv   (in place on nout)
__global__ void finalize_conv_kernel(const int* __restrict__ degn,
                                     const float* __restrict__ b_conv,
                                     float* __restrict__ nout)
{
    const int idx = blockIdx.x * blockDim.x + threadIdx.x;
    if (idx >= NN * HFD) return;
    const int n = idx / HFD, t = idx % HFD;
    const int d = degn[n];
    const float dv = (d > 0) ? 1.f / (float)d : 0.f;
    nout[idx] = nout[idx] * dv + b_conv[t];
}

// hf[m,:] = sum of conv_out over edge's 16 nodes; out[m,:] = hf @ W_out + b_out
__global__ __launch_bounds__(HFD)
void edge_out_kernel(const int* __restrict__ node_idx,
                     const float* __restrict__ conv_out,
                     const float* __restrict__ W_out,
                     const float* __restrict__ b_out,
                     float* __restrict__ out)
{
    const int m = blockIdx.x, t = threadIdx.x;
    __shared__ int nd[DEG];
    __shared__ float hf[HFD];
    if (t < DEG) nd[t] = node_idx[m * DEG + t];
    __syncthreads();
    float s = 0.f;
#pragma unroll
    for (int j = 0; j < DEG; j++) s += conv_out[(size_t)nd[j] * HFD + t];
    hf[t] = s;
    __syncthreads();
    if (t < 2) {
        float o = b_out[t];
        for (int i = 0; i < HFD; i++) o += hf[i] * W_out[i * 2 + t];
        out[m * 2 + t] = o;
    }
}

extern "C" void kernel_launch(void* const* d_in, const int* in_sizes, int n_in,
                              void* d_out, int out_size, void* d_ws, size_t ws_size,
                              hipStream_t stream)
{
    const float* feat     = (const float*)d_in[0];
    // d_in[1] = dense incidence matrix: unused (structure recovered from node_idx)
    const int*   node_idx = (const int*)  d_in[2];
    // d_in[3] = edge_idx: entries are contiguous 16-blocks per edge (sorted)
    const float* W_enc  = (const float*)d_in[4];
    const float* b_enc  = (const float*)d_in[5];
    const float* W_attr = (const float*)d_in[6];
    const float* b_attr = (const float*)d_in[7];
    const float* W_conv = (const float*)d_in[8];
    const float* att    = (const float*)d_in[9];
    const float* b_conv = (const float*)d_in[10];
    const float* W_out  = (const float*)d_in[11];
    const float* b_out  = (const float*)d_in[12];
    float* out = (float*)d_out;

    float* ws = (float*)d_ws;
    size_t o = 0;
    float* x     = ws + o; o += (size_t)NN * EMB;
    float* hattr = ws + o; o += (size_t)MM * EMB;
    float* xl    = ws + o; o += (size_t)NN * HFD;
    float* ea    = ws + o; o += (size_t)MM * HFD;
    float* eout  = ws + o; o += (size_t)MM * HFD;
    float* nout  = ws + o; o += (size_t)NN * HFD;
    float* xdot  = ws + o; o += (size_t)NN * NH;
    float* edot  = ws + o; o += (size_t)MM * NH;
    float* alpha = ws + o; o += (size_t)EE * NH;
    int*   degn  = (int*)(ws + o); o += NN;

    hipMemsetAsync(nout, 0, (size_t)NN * HFD * sizeof(float), stream);
    hipMemsetAsync(degn, 0, (size_t)NN * sizeof(int), stream);

    // x = relu(feat @ W_enc + b_enc)            [8192,512]@[512,256]
    // waves = (8192/64)*(256/32) = 1024 -> 128 blocks of 8 waves
    gemm_wmma_f32<<<dim3((NN / 64) * (EMB / 32) / 8), 256, 0, stream>>>(
        feat, W_enc, b_enc, x, FIN, EMB, 1);

    // hyperedge_attr = inc_t @ W_attr + b_attr  (16-node gather-sum)
    hedge_attr_kernel<<<MM, EMB, 0, stream>>>(node_idx, W_attr, b_attr, hattr);

    // xl = x @ W_conv ; ea = hattr @ W_conv     [8192,256]@[256,384]
    // waves = (8192/64)*(384/32) = 1536 -> 192 blocks
    gemm_wmma_f32<<<dim3((NN / 64) * (HFD / 32) / 8), 256, 0, stream>>>(
        x, W_conv, nullptr, xl, EMB, HFD, 0);
    gemm_wmma_f32<<<dim3((MM / 64) * (HFD / 32) / 8), 256, 0, stream>>>(
        hattr, W_conv, nullptr, ea, EMB, HFD, 0);

    att_dots_kernel<<<(NN * NH + 255) / 256, 256, 0, stream>>>(
        xl, ea, att, xdot, edot);

    node_deg_kernel<<<(EE + 255) / 256, 256, 0, stream>>>(node_idx, degn);

    edge_softmax_eout_kernel<<<MM, HFD, 0, stream>>>(
        node_idx, xdot, edot, xl, alpha, eout);

    scatter_nodes_kernel<<<MM, HFD, 0, stream>>>(node_idx, alpha, eout, nout);

    finalize_conv_kernel<<<(NN * HFD + 255) / 256, 256, 0, stream>>>(
        degn, b_conv, nout);

    edge_out_kernel<<<MM, HFD, 0, stream>>>(node_idx, nout, W_out, b_out, out);
}